// Model_41266045780584
// MI455X (gfx1250) — compile-verified
//
#include <hip/hip_runtime.h>
#include <math.h>

typedef float v2f __attribute__((ext_vector_type(2)));
typedef float v8f __attribute__((ext_vector_type(8)));

#define PLANE   16384                 // 128*128 floats per (b,c) plane
#define THREADS 256                   // 8 waves (wave32)
#define NWAVES  (THREADS / 32)
#define CHUNK   (PLANE / NWAVES)      // 2048 floats per wave
#define WITERS  (CHUNK / 64)          // 32 WMMA f32 16x16x4 steps per wave per pass

__device__ __forceinline__ float wave_reduce_sum(float s) {
#pragma unroll
    for (int off = 16; off >= 1; off >>= 1)
        s += __shfl_xor(s, off, 32);
    return s;
}

// One workgroup per (b,c) plane. Async-DMA the 64KiB plane into LDS once,
// then two WMMA-driven reduction passes (raw sum -> mean, clipped sum -> descriptor).
__global__ __launch_bounds__(THREADS) void plane_kernel(const float* __restrict__ x,
                                                        float* __restrict__ desc) {
    __shared__ __align__(16) float lds[PLANE];
    __shared__ float red[NWAVES];
    __shared__ float mean_sh;

    const int tid  = threadIdx.x;
    const int lane = tid & 31;
    const int wave = tid >> 5;

    const float* src = x + (size_t)blockIdx.x * PLANE;

    // --- Stage plane: global -> LDS via async tensor-path loads (ASYNCcnt) ---
    // Low 32 bits of a generic shared pointer are the LDS byte address.
    unsigned lds_base = (unsigned)(size_t)&lds[0];
    unsigned v_off    = (unsigned)tid * 16u;    // 16B per lane per round
#pragma unroll
    for (int r = 0; r < 16; ++r) {              // 256 lanes * 16B * 16 rounds = 64KiB
        unsigned off = v_off + (unsigned)r * 4096u;
        asm volatile("global_load_async_to_lds_b128 %0, %1, %2 offset:0"
                     :: "v"(lds_base + off), "v"(off), "s"(src)
                     : "memory");
    }
    asm volatile("s_wait_asynccnt 0x0" ::: "memory");
    __syncthreads();

    const float* wbase = &lds[wave * CHUNK];
    v2f ones; ones[0] = 1.0f; ones[1] = 1.0f;

    // --- Pass 1: raw sum via WMMA (D = A x ones + C). sum(C) == 16 * sum(A). ---
    v8f acc = {};
#pragma unroll 4
    for (int i = 0; i < WITERS; ++i) {
        v2f a = *(const v2f*)(wbase + i * 64 + lane * 2);
        acc = __builtin_amdgcn_wmma_f32_16x16x4_f32(
            /*neg_a=*/false, a, /*neg_b=*/false, ones,
            /*c_mod=*/(short)0, acc, /*reuse_a=*/false, /*reuse_b=*/false);
    }
    float s = acc[0] + acc[1] + acc[2] + acc[3] + acc[4] + acc[5] + acc[6] + acc[7];
    s = wave_reduce_sum(s);
    if (lane == 0) red[wave] = s;
    __syncthreads();
    if (tid == 0) {
        float t = 0.0f;
#pragma unroll
        for (int w = 0; w < NWAVES; ++w) t += red[w];
        mean_sh = t * (1.0f / (16.0f * (float)PLANE));   // /16 (WMMA fan-out) /16384
    }
    __syncthreads();
    const float mean = mean_sh;

    // --- Pass 2: sum of clip(x - mean, 0, 6) via WMMA ---
    v8f acc2 = {};
#pragma unroll 4
    for (int i = 0; i < WITERS; ++i) {
        v2f a = *(const v2f*)(wbase + i * 64 + lane * 2);
        a[0] = fminf(fmaxf(a[0] - mean, 0.0f), 6.0f);
        a[1] = fminf(fmaxf(a[1] - mean, 0.0f), 6.0f);
        acc2 = __builtin_amdgcn_wmma_f32_16x16x4_f32(
            false, a, false, ones, (short)0, acc2, false, false);
    }
    float s2 = acc2[0] + acc2[1] + acc2[2] + acc2[3] + acc2[4] + acc2[5] + acc2[6] + acc2[7];
    s2 = wave_reduce_sum(s2);
    if (lane == 0) red[wave] = s2;
    __syncthreads();
    if (tid == 0) {
        float t = 0.0f;
#pragma unroll
        for (int w = 0; w < NWAVES; ++w) t += red[w];
        desc[blockIdx.x] = t * (1.0f / (16.0f * (float)PLANE));
    }
}

// Per-sample L2 normalization over 128 channels. 16 blocks x 128 threads.
__global__ __launch_bounds__(128) void norm_kernel(const float* __restrict__ desc,
                                                   float* __restrict__ out) {
    __shared__ float red[4];
    const int b    = blockIdx.x;
    const int c    = threadIdx.x;
    const int lane = c & 31;
    const int wave = c >> 5;

    float d  = desc[b * 128 + c];
    float sq = wave_reduce_sum(d * d);
    if (lane == 0) red[wave] = sq;
    __syncthreads();
    float ss = red[0] + red[1] + red[2] + red[3];
    out[b * 128 + c] = d / sqrtf(ss + 1e-6f);
}

extern "C" void kernel_launch(void* const* d_in, const int* in_sizes, int n_in,
                              void* d_out, int out_size, void* d_ws, size_t ws_size,
                              hipStream_t stream) {
    (void)n_in; (void)out_size; (void)ws_size;
    const float* x    = (const float*)d_in[0];
    float*       desc = (float*)d_ws;          // 2048 floats of scratch
    float*       out  = (float*)d_out;

    const int planes  = in_sizes[0] / PLANE;   // 16*128 = 2048
    const int samples = planes / 128;          // 16

    plane_kernel<<<planes, THREADS, 0, stream>>>(x, desc);
    norm_kernel<<<samples, 128, 0, stream>>>(desc, out);
}